// GateMulti_71133248356698
// MI455X (gfx1250) — compile-verified
//
#include <hip/hip_runtime.h>

typedef __attribute__((ext_vector_type(2))) float v2f;
typedef __attribute__((ext_vector_type(8))) float v8f;

#define N_SUBNET 8
#define BTOK     4096
#define D_IN     512
#define D_FF     2048
#define D_OUT    512

#define BM       32     // rows per block (2 row groups of 16)
#define BN       256    // cols per block: 8 waves x 16 cols x 2 slabs (+128)
#define KCHUNK   256    // K staged in LDS per iteration
#define LDS_PAD  4      // row stride 260 floats -> bank offset 4/row, conflict-free frags

// ---------------- routing ----------------

__global__ void route_init_kernel(int* counts) {
    if (threadIdx.x < N_SUBNET) counts[threadIdx.x] = 0;
}

__global__ void route_scatter_kernel(const int* __restrict__ groups,
                                     int* __restrict__ counts,
                                     int* __restrict__ perm) {
    int b = blockIdx.x * blockDim.x + threadIdx.x;
    if (b < BTOK) {
        int e = groups[2 * b];               // GROUP_COL = 0
        int p = atomicAdd(&counts[e], 1);
        perm[e * BTOK + p] = b;
    }
}

__global__ void route_prefix_kernel(const int* __restrict__ counts,
                                    int* __restrict__ offsets) {
    if (threadIdx.x == 0) {
        int run = 0;
        for (int e = 0; e < N_SUBNET; ++e) { offsets[e] = run; run += counts[e]; }
    }
}

// ---------------- routed GEMM (+bias, +relu, +gather/scatter) ----------------
// Layer1: A = x (gathered via perm), W = W1[e] (K=512 x N=2048), out = h compact, relu
// Layer2: A = h compact, W = W2[e] (K=2048 x N=512), out scattered via perm, no relu
// Each wave: 1 B fragment feeds 2 row-group WMMAs; 2 col slabs amortize A LDS reads.

template <int K, int N, bool GATHER_IN, bool RELU, bool SCATTER_OUT>
__global__ __launch_bounds__(256) void moe_gemm_kernel(
    const float* __restrict__ A, const float* __restrict__ W,
    const float* __restrict__ bias, float* __restrict__ Out,
    const int* __restrict__ perm, const int* __restrict__ counts,
    const int* __restrict__ offsets)
{
    __shared__ float Xs[BM][KCHUNK + LDS_PAD];

    const int e   = blockIdx.z;
    const int cnt = counts[e];
    const int m0  = blockIdx.y * BM;
    if (m0 >= cnt) return;                 // block-uniform early exit (EXEC stays full)
    const int off = offsets[e];

    const int tid  = threadIdx.x;
    const int lane = tid & 31;
    const int wn   = tid >> 5;             // 8 waves -> 8 col groups of 16
    const int l16  = lane & 15;
    const int koff = lane >> 4;            // selects K pair {0,1} vs {2,3} (ISA 16x4 A layout)
    const int col0 = blockIdx.x * BN + wn * 16 + l16;   // N index (B/C/D: N = lane&15)

    const float* __restrict__ Wg = W + (size_t)e * K * N;

    v8f acc00 = {};   // rows 0-15,  cols col0
    v8f acc10 = {};   // rows 16-31, cols col0
    v8f acc01 = {};   // rows 0-15,  cols col0+128
    v8f acc11 = {};   // rows 16-31, cols col0+128

    for (int kc = 0; kc < K; kc += KCHUNK) {
        __syncthreads();
        // stage 32 gathered rows x KCHUNK into LDS (float4 loads, 8 per thread)
        for (int i = tid; i < BM * (KCHUNK / 4); i += 256) {
            int r  = i / (KCHUNK / 4);
            int c  = (i % (KCHUNK / 4)) * 4;
            int rl = m0 + r;
            if (rl >= cnt) rl = cnt - 1;   // clamp ragged tail (stores masked later)
            size_t arow = GATHER_IN ? (size_t)perm[e * BTOK + rl]
                                    : (size_t)(off + rl);
            const float4 v = *(const float4*)(A + arow * K + (size_t)(kc + c));
            Xs[r][c + 0] = v.x; Xs[r][c + 1] = v.y;
            Xs[r][c + 2] = v.z; Xs[r][c + 3] = v.w;
        }
        __syncthreads();

        const float* __restrict__ xrow0 = &Xs[l16][2 * koff];        // rows 0-15
        const float* __restrict__ xrow1 = &Xs[16 + l16][2 * koff];   // rows 16-31
        #pragma unroll 8
        for (int kk = 0; kk < KCHUNK; kk += 4) {
            // A frags: v0 = A[M][kbase + 2*koff], v1 = +1
            v2f a0 = *(const v2f*)(xrow0 + kk);
            v2f a1 = *(const v2f*)(xrow1 + kk);
            const int kg = kc + kk + 2 * koff;
            const float* wp = Wg + (size_t)kg * N + col0;
            // B frags: v0 = B[kg][col], v1 = B[kg+1][col]
            v2f b0, b1;
            b0.x = wp[0];    b0.y = wp[N];
            b1.x = wp[128];  b1.y = wp[N + 128];
            acc00 = __builtin_amdgcn_wmma_f32_16x16x4_f32(
                        false, a0, false, b0, (short)0, acc00, false, false);
            acc10 = __builtin_amdgcn_wmma_f32_16x16x4_f32(
                        false, a1, false, b0, (short)0, acc10, false, false);
            acc01 = __builtin_amdgcn_wmma_f32_16x16x4_f32(
                        false, a0, false, b1, (short)0, acc01, false, false);
            acc11 = __builtin_amdgcn_wmma_f32_16x16x4_f32(
                        false, a1, false, b1, (short)0, acc11, false, false);
        }
    }

    // epilogue: C/D layout — acc[i]: lanes0-15 -> (M=i, N=lane), lanes16-31 -> (M=8+i)
    const float bv0  = bias[e * N + col0];
    const float bv1  = bias[e * N + col0 + 128];
    const int   half = lane >> 4;
    #pragma unroll
    for (int wm = 0; wm < 2; ++wm) {
        const v8f& a0 = wm ? acc10 : acc00;
        const v8f& a1 = wm ? acc11 : acc01;
        #pragma unroll
        for (int i = 0; i < 8; ++i) {
            const int rloc = m0 + wm * 16 + half * 8 + i;
            if (rloc < cnt) {
                float v0 = a0[i] + bv0;
                float v1 = a1[i] + bv1;
                if (RELU) { v0 = fmaxf(v0, 0.f); v1 = fmaxf(v1, 0.f); }
                size_t orow = SCATTER_OUT ? (size_t)perm[e * BTOK + rloc]
                                          : (size_t)(off + rloc);
                Out[orow * N + col0]       = v0;
                Out[orow * N + col0 + 128] = v1;
            }
        }
    }
}

// ---------------- host ----------------

extern "C" void kernel_launch(void* const* d_in, const int* in_sizes, int n_in,
                              void* d_out, int out_size, void* d_ws, size_t ws_size,
                              hipStream_t stream) {
    (void)in_sizes; (void)n_in; (void)out_size; (void)ws_size;

    const float* x      = (const float*)d_in[0];   // [4096, 512]
    const int*   groups = (const int*)  d_in[1];   // [4096, 2]
    const float* W1     = (const float*)d_in[2];   // [8, 512, 2048]
    const float* b1     = (const float*)d_in[3];   // [8, 2048]
    const float* W2     = (const float*)d_in[4];   // [8, 2048, 512]
    const float* b2     = (const float*)d_in[5];   // [8, 512]
    float*       out    = (float*)d_out;           // [4096, 512]

    char* ws      = (char*)d_ws;
    int*  counts  = (int*)(ws);                               // 8 ints
    int*  offsets = (int*)(ws + 64);                          // 8 ints
    int*  perm    = (int*)(ws + 128);                         // 8*4096 ints
    float* h      = (float*)(ws + 128 + (size_t)N_SUBNET * BTOK * sizeof(int));
                                                              // [4096, 2048] compact

    route_init_kernel<<<1, 32, 0, stream>>>(counts);
    route_scatter_kernel<<<(BTOK + 255) / 256, 256, 0, stream>>>(groups, counts, perm);
    route_prefix_kernel<<<1, 32, 0, stream>>>(counts, offsets);

    dim3 g1(D_FF / BN, BTOK / BM, N_SUBNET);   // 8 x 128 x 8
    moe_gemm_kernel<D_IN, D_FF, true, true, false>
        <<<g1, 256, 0, stream>>>(x, W1, b1, h, perm, counts, offsets);

    dim3 g2(D_OUT / BN, BTOK / BM, N_SUBNET);  // 2 x 128 x 8
    moe_gemm_kernel<D_FF, D_OUT, false, false, true>
        <<<g2, 256, 0, stream>>>(h, W2, b2, out, perm, counts, offsets);
}